// GCN_14018773254535
// MI455X (gfx1250) — compile-verified
//
#include <hip/hip_runtime.h>
#include <hip/hip_bf16.h>
#include <math.h>

typedef float v2f __attribute__((ext_vector_type(2)));
typedef float v8f __attribute__((ext_vector_type(8)));

#define F_DIM 128

// ---------------------------------------------------------------------------
// Degree / normalization kernels
// ---------------------------------------------------------------------------
__global__ void deg_init_kernel(float* __restrict__ deg, int n) {
    int i = blockIdx.x * blockDim.x + threadIdx.x;
    if (i < n) deg[i] = 1.0f;  // self-loop contributes 1
}

__global__ void deg_accum_kernel(float* __restrict__ deg,
                                 const int* __restrict__ dst, int e) {
    int i = blockIdx.x * blockDim.x + threadIdx.x;
    if (i < e) atomicAdd(&deg[dst[i]], 1.0f);
}

__global__ void dinv_kernel(float* __restrict__ deg, int n) {
    int i = blockIdx.x * blockDim.x + threadIdx.x;
    if (i < n) {
        float d = deg[i];
        deg[i] = (d > 0.0f) ? rsqrtf(d) : 0.0f;  // in-place deg -> d^-1/2
    }
}

// ---------------------------------------------------------------------------
// Dense transform: C[M x n_valid] = A[M x 128] @ W[128 x n_valid]
// via V_WMMA_F32_16X16X4_F32. One workgroup = 8 waves = 128 rows.
//
// W is staged in LDS as K-PAIRED float2s: pair-row p holds, for each col n,
// the pair (W[2p][n], W[2p+1][n]) — so a B fragment is one aligned
// ds_load_b64 straight into an even VGPR pair (no repacking movs).
// Columns are XOR-16 swizzled on pair-row parity so the two half-wave
// groups (which read adjacent pair-rows) hit disjoint LDS bank sets.
// ---------------------------------------------------------------------------
template <int NT, bool RELU>
__global__ __launch_bounds__(256)
void gemm_wmma_f32(const float* __restrict__ A, const float* __restrict__ W,
                   float* __restrict__ C, int M, int n_valid, int ldc) {
    constexpr int NT16 = NT * 16;
    constexpr int LDSW = ((NT16 + 31) / 32) * 32;   // pair-row width (cols)
    __shared__ float sW[64 * LDSW * 2];             // 64 pair-rows of float2

    const int tid = threadIdx.x;
    // cooperative fill (scalar, once per block): zero-pad cols >= n_valid
    for (int idx = tid; idx < F_DIM * LDSW; idx += 256) {
        int k = idx / LDSW;            // 0..127
        int n = idx % LDSW;            // 0..LDSW-1
        float v = (n < n_valid) ? W[k * n_valid + n] : 0.0f;
        int p   = k >> 1;                              // pair-row
        int nsw = n ^ ((p & 1) << 4);                  // bank swizzle
        sW[(p * LDSW + nsw) * 2 + (k & 1)] = v;
    }
    __syncthreads();

    const int wave = tid >> 5;
    const int lane = tid & 31;
    const int half = lane >> 4;   // 0: K pair {k0,k0+1}, 1: {k0+2,k0+3}
    const int l16  = lane & 15;

    const int r0 = blockIdx.x * 128 + wave * 16;
    int row = r0 + l16;
    if (row >= M) row = M - 1;               // clamp loads; stores predicated
    const float* arow = A + (size_t)row * F_DIM;
    const float2* sW2 = (const float2*)sW;

    v8f acc[NT] = {};

    for (int k0 = 0; k0 < F_DIM; k0 += 4) {
        // A fragment (16x4 f32): lanes 0-15 hold K=k0,k0+1; lanes 16-31 K=k0+2,k0+3
        float2 av = *(const float2*)(arow + k0 + 2 * half);
        float ax = av.x, ay = av.y;
        if (RELU) { ax = fmaxf(ax, 0.0f); ay = fmaxf(ay, 0.0f); }
        v2f a; a.x = ax; a.y = ay;

        const int prow = (k0 >> 1) + half;             // pair-row for this half
        const int swz  = (prow & 1) << 4;
        const float2* bp = &sW2[prow * LDSW + l16];
#pragma unroll
        for (int t = 0; t < NT; ++t) {
            float2 bv = bp[(t * 16) ^ swz];            // single ds_load_b64
            v2f b; b.x = bv.x; b.y = bv.y;
            acc[t] = __builtin_amdgcn_wmma_f32_16x16x4_f32(
                false, a, false, b, (short)0, acc[t], false, false);
        }
    }

    // C/D layout: VGPR v -> row r0 + v + 8*half ; col = t*16 + l16
#pragma unroll
    for (int t = 0; t < NT; ++t) {
        int n = t * 16 + l16;
#pragma unroll
        for (int v = 0; v < 8; ++v) {
            int r = r0 + 8 * half + v;
            if (r < M && n < n_valid) C[(size_t)r * ldc + n] = acc[t][v];
        }
    }
}

// ---------------------------------------------------------------------------
// out[i][j] = bias[j] + h[i][j] * dinv[i]^2   (self-loop message + bias init)
// ---------------------------------------------------------------------------
__global__ void init_self_kernel(float* __restrict__ out,
                                 const float* __restrict__ h,
                                 const float* __restrict__ dinv,
                                 const float* __restrict__ bias,
                                 int n, int f) {
    int idx = blockIdx.x * blockDim.x + threadIdx.x;
    if (idx >= n * f) return;
    int i = idx / f;
    int j = idx - i * f;
    float w = dinv[i];
    out[idx] = bias[j] + h[idx] * (w * w);
}

// ---------------------------------------------------------------------------
// Edge scatter, F=128: one wave32 per edge, float4 per lane, f32 atomics.
// h and out both fit in the 192MB L2 -> atomics resolve at L2 rates.
// ---------------------------------------------------------------------------
__global__ __launch_bounds__(256)
void scatter_f128_kernel(float* __restrict__ out, const float* __restrict__ h,
                         const int* __restrict__ src, const int* __restrict__ dst,
                         const float* __restrict__ dinv, int e) {
    int gw = (int)((blockIdx.x * 256u + threadIdx.x) >> 5);
    if (gw >= e) return;
    int lane = threadIdx.x & 31;
    int s = src[gw];
    int d = dst[gw];
    float w = dinv[s] * dinv[d];
    const float4 hv = *(const float4*)(h + (size_t)s * F_DIM + lane * 4);
    float* o = out + (size_t)d * F_DIM + lane * 4;
    atomicAdd(o + 0, hv.x * w);
    atomicAdd(o + 1, hv.y * w);
    atomicAdd(o + 2, hv.z * w);
    atomicAdd(o + 3, hv.w * w);
}

// Edge scatter, F=40 (final layer)
__global__ __launch_bounds__(256)
void scatter_f40_kernel(float* __restrict__ out, const float* __restrict__ h,
                        const int* __restrict__ src, const int* __restrict__ dst,
                        const float* __restrict__ dinv, int e) {
    int gw = (int)((blockIdx.x * 256u + threadIdx.x) >> 5);
    if (gw >= e) return;
    int lane = threadIdx.x & 31;
    int s = src[gw];
    int d = dst[gw];
    float w = dinv[s] * dinv[d];
    const float* hs = h + (size_t)s * 40;
    float* o = out + (size_t)d * 40;
    for (int j = lane; j < 40; j += 32) atomicAdd(o + j, hs[j] * w);
}

// ---------------------------------------------------------------------------
// Row-wise log_softmax over 40 classes; one wave32 per row, shfl reductions.
// ---------------------------------------------------------------------------
__global__ __launch_bounds__(256)
void log_softmax40_kernel(float* __restrict__ out, int n) {
    int row = (int)((blockIdx.x * 256u + threadIdx.x) >> 5);
    if (row >= n) return;
    int lane = threadIdx.x & 31;
    float* p = out + (size_t)row * 40;

    float x0 = p[lane];                                   // lanes 0..31
    float x1 = (lane + 32 < 40) ? p[lane + 32] : -INFINITY;

    float m = fmaxf(x0, x1);
#pragma unroll
    for (int off = 16; off > 0; off >>= 1)
        m = fmaxf(m, __shfl_xor(m, off, 32));

    float s = expf(x0 - m) + ((lane + 32 < 40) ? expf(x1 - m) : 0.0f);
#pragma unroll
    for (int off = 16; off > 0; off >>= 1)
        s += __shfl_xor(s, off, 32);

    float lg = m + logf(s);
    p[lane] = x0 - lg;
    if (lane + 32 < 40) p[lane + 32] = x1 - lg;
}

// ---------------------------------------------------------------------------
// Launch
// ---------------------------------------------------------------------------
extern "C" void kernel_launch(void* const* d_in, const int* in_sizes, int n_in,
                              void* d_out, int out_size, void* d_ws, size_t ws_size,
                              hipStream_t stream) {
    const float* x  = (const float*)d_in[0];
    const int*   ei = (const int*)d_in[1];
    const float* W1 = (const float*)d_in[2];
    const float* b1 = (const float*)d_in[3];
    const float* W2 = (const float*)d_in[4];
    const float* b2 = (const float*)d_in[5];
    const float* W3 = (const float*)d_in[6];
    const float* b3 = (const float*)d_in[7];
    float* out = (float*)d_out;

    const int N = in_sizes[0] / F_DIM;   // 169343
    const int E = in_sizes[1] / 2;       // 1166243
    const int* src = ei;
    const int* dst = ei + E;

    // workspace: dinv[N] | bufA[N*128] | bufB[N*128]
    float* dinv = (float*)d_ws;
    float* bufA = dinv + (((size_t)N + 255) / 256) * 256;
    float* bufB = bufA + (size_t)N * F_DIM;

    const int TPB = 256;
    const int gN   = (N + TPB - 1) / TPB;
    const int gE   = (E + TPB - 1) / TPB;
    const int gNF  = (N * F_DIM + TPB - 1) / TPB;
    const int gN40 = (N * 40 + TPB - 1) / TPB;
    const int gGemm = (N + 127) / 128;
    const int gEdgeWave = (E + 7) / 8;   // 8 waves per 256-thread block
    const int gRowWave  = (N + 7) / 8;

    // normalization: deg -> dinv
    deg_init_kernel<<<gN, TPB, 0, stream>>>(dinv, N);
    deg_accum_kernel<<<gE, TPB, 0, stream>>>(dinv, dst, E);
    dinv_kernel<<<gN, TPB, 0, stream>>>(dinv, N);

    // ---- layer 1: H1 = x @ W1 ; A1 = b1 + scatter(H1) ----
    gemm_wmma_f32<8, false><<<gGemm, TPB, 0, stream>>>(x, W1, bufA, N, F_DIM, F_DIM);
    init_self_kernel<<<gNF, TPB, 0, stream>>>(bufB, bufA, dinv, b1, N, F_DIM);
    scatter_f128_kernel<<<gEdgeWave, TPB, 0, stream>>>(bufB, bufA, src, dst, dinv, E);

    // ---- layer 2: H2 = relu(A1) @ W2 ; A2 = b2 + scatter(H2) ----
    gemm_wmma_f32<8, true><<<gGemm, TPB, 0, stream>>>(bufB, W2, bufA, N, F_DIM, F_DIM);
    init_self_kernel<<<gNF, TPB, 0, stream>>>(bufB, bufA, dinv, b2, N, F_DIM);
    scatter_f128_kernel<<<gEdgeWave, TPB, 0, stream>>>(bufB, bufA, src, dst, dinv, E);

    // ---- layer 3: H3 = A2 @ W3 (128 -> 40, padded to 64-wide LDS rows) ----
    gemm_wmma_f32<3, false><<<gGemm, TPB, 0, stream>>>(bufB, W3, bufA, N, 40, 40);
    init_self_kernel<<<gN40, TPB, 0, stream>>>(out, bufA, dinv, b3, N, 40);
    scatter_f40_kernel<<<gEdgeWave, TPB, 0, stream>>>(out, bufA, src, dst, dinv, E);

    // log_softmax
    log_softmax40_kernel<<<gRowWave, TPB, 0, stream>>>(out, N);
}